// SparseTrendInteractor_54752243089757
// MI455X (gfx1250) — compile-verified
//
#include <hip/hip_runtime.h>
#include <math.h>

// Problem constants from the reference
#define B_ 16
#define C_ 512
#define D_ 256
#define K_ 32
#define M_TOTAL (B_ * C_)          // 8192 rows
#define LDA 260                    // LDS row stride (floats): 256 + 4 pad -> conflict-free
                                   // 16x4 f32 A-fragment b64 reads (banks 4*l+k..+3 cover all 64)

typedef float v2f __attribute__((ext_vector_type(2)));
typedef float v4f __attribute__((ext_vector_type(4)));
typedef float v8f __attribute__((ext_vector_type(8)));

// D(16x16,f32) += A(16x4,f32) * B(4x16,f32)  -- full-precision CDNA5 WMMA
__device__ __forceinline__ v8f wmma4_f32(v2f a, v2f b, v8f c) {
  return __builtin_amdgcn_wmma_f32_16x16x4_f32(
      /*neg_a=*/false, a, /*neg_b=*/false, b,
      /*c_mod=*/(short)0, c, /*reuse_a=*/false, /*reuse_b=*/false);
}

// ---------------------------------------------------------------------------
// Kernel 1: softmax over K=32 scores + attention-weighted gather of neighbors.
//   wg[row, :] = sum_k softmax(scores[row])_k * h[b, idx[row,k], :]
// One block (256 threads) per row; wave 0 does the 32-wide softmax via shuffles.
// Gather is L2-resident (each batch slice of h = 512 KB).
// ---------------------------------------------------------------------------
__global__ __launch_bounds__(256) void k_gather_softmax(
    const float* __restrict__ h, const int* __restrict__ topk_idx,
    const float* __restrict__ scores, float* __restrict__ wg) {
  __shared__ float sw[K_];
  __shared__ int   sidx[K_];
  const int row = blockIdx.x;          // 0..8191
  const int b   = row >> 9;            // row / C_
  const int t   = threadIdx.x;         // 0..255 -> column of D

  if (t < K_) {                        // threads 0..31 == wave 0 (wave32)
    float s  = scores[(size_t)row * K_ + t];
    int   id = topk_idx[(size_t)row * K_ + t];
    float m = s;
    #pragma unroll
    for (int off = 16; off >= 1; off >>= 1) m = fmaxf(m, __shfl_xor(m, off, 32));
    float e = expf(s - m);
    float sum = e;
    #pragma unroll
    for (int off = 16; off >= 1; off >>= 1) sum += __shfl_xor(sum, off, 32);
    sw[t]   = e / sum;
    sidx[t] = id;
  }
  __syncthreads();

  const float* hb = h + (size_t)b * C_ * D_;
  float acc = 0.0f;
  #pragma unroll 4
  for (int k = 0; k < K_; ++k) {
    acc = fmaf(sw[k], hb[(size_t)sidx[k] * D_ + t], acc);  // coalesced 1KB row reads
  }
  wg[(size_t)row * D_ + t] = acc;
}

// ---------------------------------------------------------------------------
// Kernel 2: agg = wg @ vW + vb    ([8192,256] x [256,256])
// Block: 256 thr = 8 waves arranged 2(M) x 4(N); block tile M=64, N=256.
// Each wave owns M=32 x N=64 (2x4 accumulator tiles) so every B fragment
// (weight) load feeds TWO WMMAs -> 1 global_load_b32 per v_wmma.
// A tile staged in LDS (stride-260 pad); vW is L2-resident.
// ---------------------------------------------------------------------------
__global__ __launch_bounds__(256) void k_agg_gemm(
    const float* __restrict__ wg, const float* __restrict__ vW,
    const float* __restrict__ vb, float* __restrict__ agg) {
  __shared__ float sA[64 * LDA];
  const int m0 = blockIdx.x * 64;
  const int t  = threadIdx.x;

  // Cooperative stage: 64x256 f32 tile, float4 vectorized (16 per thread)
  #pragma unroll
  for (int i = 0; i < 16; ++i) {
    int e4 = t + i * 256;              // float4 index within tile
    int r  = e4 >> 6;                  // 64 float4 per row
    int c4 = e4 & 63;
    *(v4f*)(&sA[r * LDA + c4 * 4]) =
        *(const v4f*)(wg + (size_t)(m0 + r) * D_ + c4 * 4);
  }
  __syncthreads();

  const int lane = t & 31, wave = t >> 5;
  const int wm = wave >> 2, wn = wave & 3;          // wm: 0..1, wn: 0..3
  const int half = lane >> 4, lm = lane & 15;
  // A fragment bases for the two 16-row M-subtiles of this wave
  const float* sAw0 = sA + (wm * 32 + lm) * LDA + 2 * half;
  const float* sAw1 = sAw0 + 16 * LDA;

  v8f acc[2][4] = {{v8f{}, v8f{}, v8f{}, v8f{}}, {v8f{}, v8f{}, v8f{}, v8f{}}};
  for (int kk = 0; kk < 64; ++kk) {    // K = 256, 4 per WMMA
    const int k0 = kk * 4;
    v2f a0 = *(const v2f*)(sAw0 + k0);
    v2f a1 = *(const v2f*)(sAw1 + k0);
    const int krow = k0 + 2 * half;
    #pragma unroll
    for (int nt = 0; nt < 4; ++nt) {
      const int n = wn * 64 + nt * 16 + lm;
      const float* w = vW + (size_t)krow * D_ + n;
      v2f bb; bb.x = w[0]; bb.y = w[D_];
      acc[0][nt] = wmma4_f32(a0, bb, acc[0][nt]);
      acc[1][nt] = wmma4_f32(a1, bb, acc[1][nt]);
    }
  }

  // Epilogue: C/D layout  m = 8*half + v, n = n0 + lm
  #pragma unroll
  for (int ms = 0; ms < 2; ++ms) {
    #pragma unroll
    for (int nt = 0; nt < 4; ++nt) {
      const int n = wn * 64 + nt * 16 + lm;
      const float bias = vb[n];
      const int mrow = m0 + wm * 32 + ms * 16 + half * 8;
      #pragma unroll
      for (int v = 0; v < 8; ++v)
        agg[(size_t)(mrow + v) * D_ + n] = acc[ms][nt][v] + bias;
    }
  }
}

// ---------------------------------------------------------------------------
// Kernel 3: delta = sigmoid(h@gW[:256] + agg@gW[256:] + gb) * (agg@oW + ob)
// Same tiling as kernel 2; the 3 weight fragments per N-tile per K-step feed
// 6 WMMAs (2 M-subtiles) -> 1 load per v_wmma.
// ---------------------------------------------------------------------------
__global__ __launch_bounds__(256) void k_gate_out(
    const float* __restrict__ h, const float* __restrict__ agg,
    const float* __restrict__ gW, const float* __restrict__ gb,
    const float* __restrict__ oW, const float* __restrict__ ob,
    float* __restrict__ out) {
  __shared__ float sH[64 * LDA];
  __shared__ float sG[64 * LDA];
  const int m0 = blockIdx.x * 64;
  const int t  = threadIdx.x;

  #pragma unroll
  for (int i = 0; i < 16; ++i) {
    int e4 = t + i * 256;
    int r  = e4 >> 6;
    int c4 = e4 & 63;
    *(v4f*)(&sH[r * LDA + c4 * 4]) =
        *(const v4f*)(h + (size_t)(m0 + r) * D_ + c4 * 4);
    *(v4f*)(&sG[r * LDA + c4 * 4]) =
        *(const v4f*)(agg + (size_t)(m0 + r) * D_ + c4 * 4);
  }
  __syncthreads();

  const int lane = t & 31, wave = t >> 5;
  const int wm = wave >> 2, wn = wave & 3;
  const int half = lane >> 4, lm = lane & 15;
  const float* sHw0 = sH + (wm * 32 + lm) * LDA + 2 * half;
  const float* sHw1 = sHw0 + 16 * LDA;
  const float* sGw0 = sG + (wm * 32 + lm) * LDA + 2 * half;
  const float* sGw1 = sGw0 + 16 * LDA;

  v8f accG[2][4] = {{v8f{}, v8f{}, v8f{}, v8f{}}, {v8f{}, v8f{}, v8f{}, v8f{}}};
  v8f accO[2][4] = {{v8f{}, v8f{}, v8f{}, v8f{}}, {v8f{}, v8f{}, v8f{}, v8f{}}};
  for (int kk = 0; kk < 64; ++kk) {
    const int k0 = kk * 4;
    v2f aH0 = *(const v2f*)(sHw0 + k0);
    v2f aH1 = *(const v2f*)(sHw1 + k0);
    v2f aA0 = *(const v2f*)(sGw0 + k0);
    v2f aA1 = *(const v2f*)(sGw1 + k0);
    const int krow = k0 + 2 * half;
    #pragma unroll
    for (int nt = 0; nt < 4; ++nt) {
      const int n = wn * 64 + nt * 16 + lm;
      const float* wgt = gW + (size_t)krow * D_ + n;          // gW rows [0,256) : h
      const float* wgb = gW + (size_t)(256 + krow) * D_ + n;  // gW rows [256,512): agg
      const float* wo  = oW + (size_t)krow * D_ + n;
      v2f b1; b1.x = wgt[0]; b1.y = wgt[D_];
      v2f b2; b2.x = wgb[0]; b2.y = wgb[D_];
      v2f b3; b3.x = wo[0];  b3.y = wo[D_];
      accG[0][nt] = wmma4_f32(aH0, b1, accG[0][nt]);
      accG[1][nt] = wmma4_f32(aH1, b1, accG[1][nt]);
      accG[0][nt] = wmma4_f32(aA0, b2, accG[0][nt]);
      accG[1][nt] = wmma4_f32(aA1, b2, accG[1][nt]);
      accO[0][nt] = wmma4_f32(aA0, b3, accO[0][nt]);
      accO[1][nt] = wmma4_f32(aA1, b3, accO[1][nt]);
    }
  }

  #pragma unroll
  for (int ms = 0; ms < 2; ++ms) {
    #pragma unroll
    for (int nt = 0; nt < 4; ++nt) {
      const int n = wn * 64 + nt * 16 + lm;
      const float bg = gb[n], bo = ob[n];
      const int mrow = m0 + wm * 32 + ms * 16 + half * 8;
      #pragma unroll
      for (int v = 0; v < 8; ++v) {
        const float g = accG[ms][nt][v] + bg;
        const float o = accO[ms][nt][v] + bo;
        const float sg = 1.0f / (1.0f + expf(-g));
        out[(size_t)(mrow + v) * D_ + n] = sg * o;
      }
    }
  }
}

// ---------------------------------------------------------------------------
extern "C" void kernel_launch(void* const* d_in, const int* in_sizes, int n_in,
                              void* d_out, int out_size, void* d_ws, size_t ws_size,
                              hipStream_t stream) {
  const float* h      = (const float*)d_in[0];
  const int*   idx    = (const int*)d_in[1];
  const float* scores = (const float*)d_in[2];
  const float* vW     = (const float*)d_in[3];
  const float* vb     = (const float*)d_in[4];
  const float* oW     = (const float*)d_in[5];
  const float* ob     = (const float*)d_in[6];
  const float* gW     = (const float*)d_in[7];
  const float* gb     = (const float*)d_in[8];
  float* out = (float*)d_out;

  // wg (weighted gather) is staged in d_out (overwritten by kernel 3 last);
  // agg lives in the workspace (8 MB).
  float* wg  = out;
  float* agg = (float*)d_ws;

  k_gather_softmax<<<M_TOTAL, 256, 0, stream>>>(h, idx, scores, wg);
  k_agg_gemm<<<M_TOTAL / 64, 256, 0, stream>>>(wg, vW, vb, agg);
  k_gate_out<<<M_TOTAL / 64, 256, 0, stream>>>(h, agg, gW, gb, oW, ob, out);
}